// Net_21887153340561
// MI455X (gfx1250) — compile-verified
//
#include <hip/hip_runtime.h>
#include <math.h>
#include <float.h>
#include <limits.h>

#define N_NODES   100000
#define N_EDGES   1600000
#define N_FEAT    16
#define N_GRAPHS  256
#define N_CLASSES 10

typedef __attribute__((ext_vector_type(2))) float v2f;
typedef __attribute__((ext_vector_type(8))) float v8f;

// ---------- helpers ----------
__device__ __forceinline__ unsigned ord_f32(float f) {
    unsigned u = __float_as_uint(f);
    return (u & 0x80000000u) ? ~u : (u | 0x80000000u);
}
__device__ __forceinline__ float unord_f32(unsigned u) {
    return (u & 0x80000000u) ? __uint_as_float(u & 0x7FFFFFFFu)
                             : __uint_as_float(~u);
}

// ---------- generic fills ----------
__global__ void fill_u32(unsigned* p, size_t n, unsigned v) {
    size_t i = (size_t)blockIdx.x * blockDim.x + threadIdx.x;
    if (i < n) p[i] = v;
}

// ---------- degrees ----------
__global__ void deg_kernel(const int* __restrict__ row, const int* __restrict__ col,
                           float* __restrict__ deg_tgt, float* __restrict__ deg_src) {
    int e = blockIdx.x * blockDim.x + threadIdx.x;
    if (e >= N_EDGES) return;
    atomicAdd(&deg_tgt[col[e]], 1.0f);
    atomicAdd(&deg_src[row[e]], 1.0f);
}

__global__ void dis_kernel(const float* __restrict__ deg_tgt, const float* __restrict__ deg_src,
                           float* __restrict__ dis, float* __restrict__ inv_deg) {
    int i = blockIdx.x * blockDim.x + threadIdx.x;
    if (i >= N_NODES) return;
    dis[i] = rsqrtf(fmaxf(deg_tgt[i], 1.0f));       // deg includes self-loop (>=1)
    float d = deg_src[i];
    inv_deg[i] = (d > 0.0f) ? (1.0f / fmaxf(d, 1.0f)) : 0.0f;
}

// ---------- fp32 WMMA GEMM: Y[M,Nout] = X[M,K] @ W[K,Nout] ----------
// One wave32 per 16x16 output tile; V_WMMA_F32_16X16X4_F32 over K in steps of 4.
__global__ void gemm_wmma_f32(const float* __restrict__ X, const float* __restrict__ W,
                              float* __restrict__ Y, int K, int Nout) {
    int tm   = blockIdx.x;             // M tile
    int tn   = blockIdx.y;             // N tile
    int lane = threadIdx.x & 31;
    int half = lane >> 4;              // 0: K=0,1 ; 1: K=2,3 (per ISA A/B fp32 layout)
    int l16  = lane & 15;

    const float* Xrow = X + (size_t)(tm * 16 + l16) * K;   // A: lane holds row M=l16
    int ncol = tn * 16 + l16;                              // B/C: lane holds col N=l16

    v8f acc = {};
    for (int k = 0; k < K; k += 4) {
        v2f a, b;
        a.x = Xrow[k + 2 * half];
        a.y = Xrow[k + 2 * half + 1];
        b.x = W[(size_t)(k + 2 * half)     * Nout + ncol];
        b.y = W[(size_t)(k + 2 * half + 1) * Nout + ncol];
        acc = __builtin_amdgcn_wmma_f32_16x16x4_f32(
            /*neg_a=*/false, a, /*neg_b=*/false, b,
            /*c_mod=*/(short)0, acc, /*reuse_a=*/false, /*reuse_b=*/false);
    }
    // C/D layout: VGPR r -> row M = r + 8*half, col N = l16
    float* Yt = Y + (size_t)(tm * 16 + 8 * half) * Nout + ncol;
    #pragma unroll
    for (int r = 0; r < 8; ++r) Yt[(size_t)r * Nout] = acc[r];
}

// ---------- GCN aggregation ----------
// self-loop term: out[i][f] = dis[i]^2 * xw[i][f]
template <int F>
__global__ void self_init(const float* __restrict__ xw, const float* __restrict__ dis,
                          float* __restrict__ out) {
    size_t i = (size_t)blockIdx.x * blockDim.x + threadIdx.x;
    if (i >= (size_t)N_NODES * F) return;
    int node = (int)(i / F);
    float d = dis[node];
    out[i] = d * d * xw[i];
}

// edge term: out[col] += dis[row]*dis[col] * xw[row]
template <int F>
__global__ void edge_agg(const int* __restrict__ row, const int* __restrict__ col,
                         const float* __restrict__ dis, const float* __restrict__ xw,
                         float* __restrict__ out) {
    int e = blockIdx.x * blockDim.x + threadIdx.x;
    if (e >= N_EDGES) return;
    int r = row[e], c = col[e];
    float nrm = dis[r] * dis[c];
    const float* src = xw + (size_t)r * F;
    float* dst = out + (size_t)c * F;
    #pragma unroll
    for (int f = 0; f < F; ++f) atomicAdd(&dst[f], nrm * src[f]);
}

template <int F, bool RELU>
__global__ void add_bias(float* __restrict__ h, const float* __restrict__ b) {
    size_t i = (size_t)blockIdx.x * blockDim.x + threadIdx.x;
    if (i >= (size_t)N_NODES * F) return;
    float v = h[i] + b[i % F];
    h[i] = RELU ? fmaxf(v, 0.0f) : v;
}

// ---------- normalized cut + graclus ----------
__global__ void cut_kernel(const int* __restrict__ row, const int* __restrict__ col,
                           const float* __restrict__ pos, const float* __restrict__ inv_deg,
                           float* __restrict__ cut, unsigned* __restrict__ maxw_u) {
    int e = blockIdx.x * blockDim.x + threadIdx.x;
    if (e >= N_EDGES) return;
    int r = row[e], c = col[e];
    float dx = pos[2 * r] - pos[2 * c];
    float dy = pos[2 * r + 1] - pos[2 * c + 1];
    float w = sqrtf(dx * dx + dy * dy) * (inv_deg[r] + inv_deg[c]);
    cut[e] = w;
    atomicMax(&maxw_u[r], __float_as_uint(w));   // w >= 0: raw bits order == float order
}

__global__ void partner_kernel(const int* __restrict__ row, const int* __restrict__ col,
                               const float* __restrict__ cut, const unsigned* __restrict__ maxw_u,
                               int* __restrict__ partner) {
    int e = blockIdx.x * blockDim.x + threadIdx.x;
    if (e >= N_EDGES) return;
    int r = row[e];
    if (cut[e] >= __uint_as_float(maxw_u[r])) atomicMax(&partner[r], col[e]);
}

__global__ void cluster_kernel(const int* __restrict__ partner, int* __restrict__ cluster) {
    int i = blockIdx.x * blockDim.x + threadIdx.x;
    if (i >= N_NODES) return;
    int p = partner[i];
    int c = i;
    if (p >= 0 && p != i && partner[p] == i) c = min(i, p);
    cluster[i] = c;
}

// ---------- cluster max-pool (signed float via order-preserving uint) ----------
__global__ void pool_kernel(const int* __restrict__ cluster, const int* __restrict__ batch,
                            const float* __restrict__ h2, unsigned* __restrict__ xp_u,
                            int* __restrict__ cnt, int* __restrict__ bp) {
    int i = blockIdx.x * blockDim.x + threadIdx.x;
    if (i >= N_NODES) return;
    int c = cluster[i];
    atomicAdd(&cnt[c], 1);
    atomicMin(&bp[c], batch[i]);
    const float* src = h2 + (size_t)i * 64;
    unsigned* dst = xp_u + (size_t)c * 64;
    #pragma unroll 8
    for (int f = 0; f < 64; ++f) atomicMax(&dst[f], ord_f32(src[f]));
}

// ---------- global mean pool ----------
__global__ void graph_pool(const int* __restrict__ cnt, const int* __restrict__ bp,
                           const unsigned* __restrict__ xp_u,
                           float* __restrict__ sums, float* __restrict__ counts) {
    int c = blockIdx.x * blockDim.x + threadIdx.x;
    if (c >= N_NODES) return;
    if (cnt[c] <= 0) return;                 // empty clusters contribute nothing
    int b = bp[c];
    atomicAdd(&counts[b], 1.0f);
    const unsigned* src = xp_u + (size_t)c * 64;
    float* dst = sums + (size_t)b * 64;
    #pragma unroll 8
    for (int f = 0; f < 64; ++f) atomicAdd(&dst[f], unord_f32(src[f]));
}

// ---------- FC + log_softmax ----------
__global__ void fc_kernel(const float* __restrict__ sums, const float* __restrict__ counts,
                          const float* __restrict__ Wfc, const float* __restrict__ bfc,
                          float* __restrict__ out) {
    int g = blockIdx.x * blockDim.x + threadIdx.x;
    if (g >= N_GRAPHS) return;
    float inv = 1.0f / fmaxf(counts[g], 1.0f);
    float logits[N_CLASSES];
    #pragma unroll
    for (int c = 0; c < N_CLASSES; ++c) logits[c] = bfc[c];
    for (int f = 0; f < 64; ++f) {
        float m = sums[g * 64 + f] * inv;
        #pragma unroll
        for (int c = 0; c < N_CLASSES; ++c) logits[c] += m * Wfc[f * N_CLASSES + c];
    }
    float mx = logits[0];
    #pragma unroll
    for (int c = 1; c < N_CLASSES; ++c) mx = fmaxf(mx, logits[c]);
    float s = 0.0f;
    #pragma unroll
    for (int c = 0; c < N_CLASSES; ++c) s += expf(logits[c] - mx);
    float lse = logf(s);
    #pragma unroll
    for (int c = 0; c < N_CLASSES; ++c) out[g * N_CLASSES + c] = logits[c] - mx - lse;
}

// ---------- host ----------
static inline size_t rup(size_t b) { return (b + 255) & ~(size_t)255; }

extern "C" void kernel_launch(void* const* d_in, const int* in_sizes, int n_in,
                              void* d_out, int out_size, void* d_ws, size_t ws_size,
                              hipStream_t stream) {
    (void)in_sizes; (void)n_in; (void)out_size; (void)ws_size;
    const float* x    = (const float*)d_in[0];                 // [N,16]
    const int*   ei   = (const int*)d_in[1];                   // [2,E]
    const float* pos  = (const float*)d_in[2];                 // [N,2]
    const int*   bat  = (const int*)d_in[3];                   // [N]
    const float* W1   = (const float*)d_in[4];                 // [16,32]
    const float* b1   = (const float*)d_in[5];                 // [32]
    const float* W2   = (const float*)d_in[6];                 // [32,64]
    const float* b2   = (const float*)d_in[7];                 // [64]
    const float* Wfc  = (const float*)d_in[8];                 // [64,10]
    const float* bfc  = (const float*)d_in[9];                 // [10]
    float* out = (float*)d_out;

    const int* row = ei;
    const int* col = ei + N_EDGES;

    // workspace layout
    char* w = (char*)d_ws;
    auto alloc = [&](size_t bytes) -> void* { void* r = (void*)w; w += rup(bytes); return r; };
    float*    deg_tgt = (float*)alloc(N_NODES * 4);
    float*    deg_src = (float*)alloc(N_NODES * 4);
    float*    dis     = (float*)alloc(N_NODES * 4);
    float*    inv_deg = (float*)alloc(N_NODES * 4);
    float*    xw1     = (float*)alloc((size_t)N_NODES * 32 * 4);
    float*    h1      = (float*)alloc((size_t)N_NODES * 32 * 4);
    float*    xw2     = (float*)alloc((size_t)N_NODES * 64 * 4);
    float*    h2      = (float*)alloc((size_t)N_NODES * 64 * 4);
    float*    cut     = (float*)alloc((size_t)N_EDGES * 4);
    unsigned* maxw_u  = (unsigned*)alloc(N_NODES * 4);
    int*      partner = (int*)alloc(N_NODES * 4);
    int*      cluster = (int*)alloc(N_NODES * 4);
    int*      cnt     = (int*)alloc(N_NODES * 4);
    int*      bp      = (int*)alloc(N_NODES * 4);
    unsigned* xp_u    = (unsigned*)alloc((size_t)N_NODES * 64 * 4);
    float*    sums    = (float*)alloc((size_t)N_GRAPHS * 64 * 4);
    float*    counts  = (float*)alloc(N_GRAPHS * 4);

    const int T = 256;
    auto blk = [](size_t n, int t) { return (unsigned)((n + t - 1) / t); };

    // ---- init accumulators (fresh every call) ----
    fill_u32<<<blk(N_NODES, T), T, 0, stream>>>((unsigned*)deg_tgt, N_NODES, __builtin_bit_cast(unsigned, 1.0f));
    fill_u32<<<blk(N_NODES, T), T, 0, stream>>>((unsigned*)deg_src, N_NODES, 0u);
    fill_u32<<<blk(N_NODES, T), T, 0, stream>>>(maxw_u, N_NODES, 0u);
    fill_u32<<<blk(N_NODES, T), T, 0, stream>>>((unsigned*)partner, N_NODES, 0xFFFFFFFFu); // -1
    fill_u32<<<blk(N_NODES, T), T, 0, stream>>>((unsigned*)cnt, N_NODES, 0u);
    fill_u32<<<blk(N_NODES, T), T, 0, stream>>>((unsigned*)bp, N_NODES, (unsigned)INT_MAX);
    fill_u32<<<blk((size_t)N_NODES * 64, T), T, 0, stream>>>(xp_u, (size_t)N_NODES * 64, 0u);
    fill_u32<<<blk((size_t)N_GRAPHS * 64, T), T, 0, stream>>>((unsigned*)sums, (size_t)N_GRAPHS * 64, 0u);
    fill_u32<<<blk(N_GRAPHS, T), T, 0, stream>>>((unsigned*)counts, N_GRAPHS, 0u);

    // ---- degrees / norms ----
    deg_kernel<<<blk(N_EDGES, T), T, 0, stream>>>(row, col, deg_tgt, deg_src);
    dis_kernel<<<blk(N_NODES, T), T, 0, stream>>>(deg_tgt, deg_src, dis, inv_deg);

    // ---- layer 1: xw1 = x @ W1 (WMMA f32), aggregate, +b1, relu ----
    gemm_wmma_f32<<<dim3(N_NODES / 16, 32 / 16), 32, 0, stream>>>(x, W1, xw1, N_FEAT, 32);
    self_init<32><<<blk((size_t)N_NODES * 32, T), T, 0, stream>>>(xw1, dis, h1);
    edge_agg<32><<<blk(N_EDGES, T), T, 0, stream>>>(row, col, dis, xw1, h1);
    add_bias<32, true><<<blk((size_t)N_NODES * 32, T), T, 0, stream>>>(h1, b1);

    // ---- layer 2: xw2 = h1 @ W2 (WMMA f32), aggregate, +b2 ----
    gemm_wmma_f32<<<dim3(N_NODES / 16, 64 / 16), 32, 0, stream>>>(h1, W2, xw2, 32, 64);
    self_init<64><<<blk((size_t)N_NODES * 64, T), T, 0, stream>>>(xw2, dis, h2);
    edge_agg<64><<<blk(N_EDGES, T), T, 0, stream>>>(row, col, dis, xw2, h2);
    add_bias<64, false><<<blk((size_t)N_NODES * 64, T), T, 0, stream>>>(h2, b2);

    // ---- normalized cut + graclus matching ----
    cut_kernel<<<blk(N_EDGES, T), T, 0, stream>>>(row, col, pos, inv_deg, cut, maxw_u);
    partner_kernel<<<blk(N_EDGES, T), T, 0, stream>>>(row, col, cut, maxw_u, partner);
    cluster_kernel<<<blk(N_NODES, T), T, 0, stream>>>(partner, cluster);

    // ---- cluster max-pool, graph mean-pool ----
    pool_kernel<<<blk(N_NODES, T), T, 0, stream>>>(cluster, bat, h2, xp_u, cnt, bp);
    graph_pool<<<blk(N_NODES, T), T, 0, stream>>>(cnt, bp, xp_u, sums, counts);

    // ---- FC + log_softmax ----
    fc_kernel<<<blk(N_GRAPHS, T), T, 0, stream>>>(sums, counts, Wfc, bfc, out);
}